// ControlFlowScanCDist2_22445499089117
// MI455X (gfx1250) — compile-verified
//
#include <hip/hip_runtime.h>
#include <math.h>

typedef __attribute__((ext_vector_type(2))) float v2f;
typedef __attribute__((ext_vector_type(8))) float v8f;

#define N 8192
#define KDIM 64

// ---------------------------------------------------------------------------
// Prologue: per-row squared norms into workspace (8192 floats = 32 KB).
// One thread per row; 16 aligned float4 loads each. Total traffic ~2 MB.
// ---------------------------------------------------------------------------
__global__ __launch_bounds__(256) void sqnorm_kernel(const float* __restrict__ x,
                                                     float* __restrict__ sq) {
    int row = blockIdx.x * 256 + threadIdx.x;
    const float4* p = reinterpret_cast<const float4*>(x + (size_t)row * KDIM);
    float s = 0.0f;
#pragma unroll
    for (int i = 0; i < KDIM / 4; ++i) {
        float4 v = p[i];
        s += v.x * v.x + v.y * v.y + v.z * v.z + v.w * v.w;
    }
    sq[row] = s;
}

// ---------------------------------------------------------------------------
// Main kernel: each wave computes a 32x32 output super-tile as 2x2 WMMA
// 16x16 tiles using V_WMMA_F32_16X16X4_F32 (exact fp32 accumulation).
// Epilogue applies the norm-expansion + sqrt and streams results to HBM
// with non-temporal stores (output 256 MB > 192 MB L2, never re-read).
// ---------------------------------------------------------------------------
__global__ __launch_bounds__(256) void cdist_wmma_kernel(const float* __restrict__ x,
                                                         const float* __restrict__ sq,
                                                         float* __restrict__ out) {
    const int lane = threadIdx.x & 31;
    const int wid  = blockIdx.x * 8 + (threadIdx.x >> 5);   // 8 waves / block
    const int tiles = N / 32;                               // 256 per dim
    const int ti = wid / tiles;
    const int tj = wid % tiles;

    const int m  = lane & 15;      // row within 16x16 tile (A) / col (B,C,D)
    const int hi = lane >> 4;      // half-wave selector

    // A rows for the two row-tiles; B "rows" are columns of the gram (x itself).
    const float* arow0 = x + (size_t)(ti * 32 + m) * KDIM;
    const float* arow1 = arow0 + (size_t)16 * KDIM;
    const float* brow0 = x + (size_t)(tj * 32 + m) * KDIM;
    const float* brow1 = brow0 + (size_t)16 * KDIM;

    v8f acc00 = {}, acc01 = {}, acc10 = {}, acc11 = {};

#pragma unroll
    for (int kk = 0; kk < KDIM; kk += 4) {
        // f32 16x4 A layout: lanes 0-15 hold K=kk,kk+1; lanes 16-31 hold K=kk+2,kk+3
        const int kb = kk + 2 * hi;
        v2f a0 = *(const v2f*)(arow0 + kb);
        v2f a1 = *(const v2f*)(arow1 + kb);
        v2f b0 = *(const v2f*)(brow0 + kb);
        v2f b1 = *(const v2f*)(brow1 + kb);
        acc00 = __builtin_amdgcn_wmma_f32_16x16x4_f32(false, a0, false, b0,
                                                      (short)0, acc00, false, false);
        acc01 = __builtin_amdgcn_wmma_f32_16x16x4_f32(false, a0, false, b1,
                                                      (short)0, acc01, false, false);
        acc10 = __builtin_amdgcn_wmma_f32_16x16x4_f32(false, a1, false, b0,
                                                      (short)0, acc10, false, false);
        acc11 = __builtin_amdgcn_wmma_f32_16x16x4_f32(false, a1, false, b1,
                                                      (short)0, acc11, false, false);
    }

    // Epilogue: d(i,j) = sqrt(max(||xi||^2 + ||xj||^2 - 2*gram, 0))
    const int gc0 = tj * 32 + m;        // N index for col-tile 0
    const int gc1 = gc0 + 16;           // N index for col-tile 1
    const float sb0 = sq[gc0];
    const float sb1 = sq[gc1];

#pragma unroll
    for (int v = 0; v < 8; ++v) {
        // C/D layout: VGPR v, lanes 0-15 -> M=v; lanes 16-31 -> M=v+8
        const int gr0 = ti * 32 + v + 8 * hi;   // row-tile 0
        const int gr1 = gr0 + 16;               // row-tile 1
        const float sa0 = sq[gr0];
        const float sa1 = sq[gr1];

        float d;
        d = sa0 + sb0 - 2.0f * acc00[v];
        __builtin_nontemporal_store(sqrtf(fmaxf(d, 0.0f)), out + (size_t)gr0 * N + gc0);
        d = sa0 + sb1 - 2.0f * acc01[v];
        __builtin_nontemporal_store(sqrtf(fmaxf(d, 0.0f)), out + (size_t)gr0 * N + gc1);
        d = sa1 + sb0 - 2.0f * acc10[v];
        __builtin_nontemporal_store(sqrtf(fmaxf(d, 0.0f)), out + (size_t)gr1 * N + gc0);
        d = sa1 + sb1 - 2.0f * acc11[v];
        __builtin_nontemporal_store(sqrtf(fmaxf(d, 0.0f)), out + (size_t)gr1 * N + gc1);
    }
}

extern "C" void kernel_launch(void* const* d_in, const int* in_sizes, int n_in,
                              void* d_out, int out_size, void* d_ws, size_t ws_size,
                              hipStream_t stream) {
    const float* x  = (const float*)d_in[0];
    float* out      = (float*)d_out;
    float* sq       = (float*)d_ws;     // 8192 floats = 32 KB scratch

    sqnorm_kernel<<<N / 256, 256, 0, stream>>>(x, sq);

    // (8192/32)^2 = 65536 wave-jobs, 8 waves per 256-thread block -> 8192 blocks
    const int tiles = N / 32;
    cdist_wmma_kernel<<<(tiles * tiles) / 8, 256, 0, stream>>>(x, sq, out);
}